// GraphRetrieval_71743133712712
// MI455X (gfx1250) — compile-verified
//
#include <hip/hip_runtime.h>

// MI455X / gfx1250, wave32. f32 WMMA path (V_WMMA_F32_16X16X4_F32).
typedef __attribute__((ext_vector_type(2))) float v2f;
typedef __attribute__((ext_vector_type(8))) float v8f;

#define B_ROWS 16384
#define KRET   8
#define DDIM   128
#define CDIM   128
#define PITCH  132           // 16x128 tile padded: bank-conflict-free column reads
#define WAVES  4
#define ROWS_PER_BLOCK (WAVES * 16)

__global__ __launch_bounds__(WAVES * 32)
void graph_retrieval_kernel(const float* __restrict__ g,      // [B, D]
                            const float* __restrict__ retr,   // [B, K, D]
                            const float* __restrict__ Wa,     // [D, D]
                            const float* __restrict__ Wp,     // [D, C]
                            const float* __restrict__ bp,     // [C]
                            const int*   __restrict__ labels, // [B, K]
                            float* __restrict__ out)          // [B, C]
{
    __shared__ float tile[WAVES][16 * PITCH];   // graph tile, later reused for t
    __shared__ float attnS[WAVES][16 * 9];      // per-row attention weights

    const int tid  = threadIdx.x;
    const int wave = tid >> 5;
    const int lane = tid & 31;
    const int half = lane >> 4;       // 0: lanes 0-15, 1: lanes 16-31
    const int l15  = lane & 15;
    const int t0   = half * 2;        // K offset of this half within a 4-step
    const int rowBase = blockIdx.x * ROWS_PER_BLOCK + wave * 16;

    float* T = tile[wave];

    // ---- Phase 0: stage 16x128 graph tile into LDS (coalesced float4) ----
    {
        const float4* gsrc = reinterpret_cast<const float4*>(g + (size_t)rowBase * DDIM);
        for (int r = 0; r < 16; ++r) {
            float4 v = gsrc[r * 32 + lane];
            *reinterpret_cast<float4*>(&T[r * PITCH + lane * 4]) = v;
        }
    }
    __syncthreads();

    // ---- Phase 1: fused dual GEMM  logits = g@Wp,  t = g@Wa  (K=128, steps of 4) ----
    const v8f zero8 = {0.f, 0.f, 0.f, 0.f, 0.f, 0.f, 0.f, 0.f};
    v8f accP[8], accA[8];
#pragma unroll
    for (int n = 0; n < 8; ++n) { accP[n] = zero8; accA[n] = zero8; }

    for (int kk = 0; kk < DDIM; kk += 4) {
        // A fragment (16x4 f32): lanes 0-15 -> K=kk,kk+1 ; lanes 16-31 -> K=kk+2,kk+3
        v2f a;
        a.x = T[l15 * PITCH + kk + t0];
        a.y = T[l15 * PITCH + kk + t0 + 1];

        const float* wp0 = Wp + (size_t)(kk + t0) * CDIM + l15;
        const float* wa0 = Wa + (size_t)(kk + t0) * DDIM + l15;

#pragma unroll
        for (int n = 0; n < 8; ++n) {     // 8 N-tiles of W_pred
            v2f b;
            b.x = wp0[n * 16];            // row kk+t0
            b.y = wp0[CDIM + n * 16];     // row kk+t0+1
            accP[n] = __builtin_amdgcn_wmma_f32_16x16x4_f32(
                false, a, false, b, (short)0, accP[n], false, false);
        }
#pragma unroll
        for (int n = 0; n < 8; ++n) {     // 8 N-tiles of W_adapter
            v2f b;
            b.x = wa0[n * 16];
            b.y = wa0[DDIM + n * 16];
            accA[n] = __builtin_amdgcn_wmma_f32_16x16x4_f32(
                false, a, false, b, (short)0, accA[n], false, false);
        }
    }

    // ---- Phase 2: bias + row softmax of logits, in WMMA register layout ----
    // accP[n][j] holds logits[row = j + half*8][col = n*16 + l15]; each row lives
    // in 16 lanes (one half) -> reduce with shfl_xor 1,2,4,8.
    float bpv[8];
#pragma unroll
    for (int n = 0; n < 8; ++n) bpv[n] = bp[n * 16 + l15];
#pragma unroll
    for (int n = 0; n < 8; ++n)
#pragma unroll
        for (int j = 0; j < 8; ++j) accP[n][j] += bpv[n];

#pragma unroll
    for (int j = 0; j < 8; ++j) {
        float mx = -3.402823466e38f;
#pragma unroll
        for (int n = 0; n < 8; ++n) mx = fmaxf(mx, accP[n][j]);
        mx = fmaxf(mx, __shfl_xor(mx, 1, 32));
        mx = fmaxf(mx, __shfl_xor(mx, 2, 32));
        mx = fmaxf(mx, __shfl_xor(mx, 4, 32));
        mx = fmaxf(mx, __shfl_xor(mx, 8, 32));
        float s = 0.f;
#pragma unroll
        for (int n = 0; n < 8; ++n) {
            float e = __expf(accP[n][j] - mx);
            accP[n][j] = e;
            s += e;
        }
        s += __shfl_xor(s, 1, 32);
        s += __shfl_xor(s, 2, 32);
        s += __shfl_xor(s, 4, 32);
        s += __shfl_xor(s, 8, 32);
        float inv = 1.0f / s;
#pragma unroll
        for (int n = 0; n < 8; ++n) accP[n][j] *= inv;   // accP now = g_label
    }

    // ---- Phase 3: write t (= g@Wa) over the LDS tile ----
#pragma unroll
    for (int n = 0; n < 8; ++n)
#pragma unroll
        for (int j = 0; j < 8; ++j)
            T[(j + half * 8) * PITCH + n * 16 + l15] = accA[n][j];
    __syncthreads();

    // ---- Phase 4: bilinear scores + softmax over K+1=9 candidates ----
    // Row-sequential; 32 lanes cooperate on one 128-dot (coalesced float4 loads).
    for (int r = 0; r < 16; ++r) {
        const int rowG = rowBase + r;
        float4 tv = *reinterpret_cast<const float4*>(&T[r * PITCH + lane * 4]);
        float sc[9];
        {   // m = 0: concat_H[:,0] = graph embedding (re-read from global; L2-hot)
            float4 g4 = reinterpret_cast<const float4*>(g + (size_t)rowG * DDIM)[lane];
            float p = tv.x * g4.x + tv.y * g4.y + tv.z * g4.z + tv.w * g4.w;
            p += __shfl_xor(p, 1, 32);
            p += __shfl_xor(p, 2, 32);
            p += __shfl_xor(p, 4, 32);
            p += __shfl_xor(p, 8, 32);
            p += __shfl_xor(p, 16, 32);
            sc[0] = p;
        }
#pragma unroll
        for (int m = 1; m <= 8; ++m) {
            float4 r4 = reinterpret_cast<const float4*>(
                retr + ((size_t)rowG * KRET + (m - 1)) * DDIM)[lane];
            float p = tv.x * r4.x + tv.y * r4.y + tv.z * r4.z + tv.w * r4.w;
            p += __shfl_xor(p, 1, 32);
            p += __shfl_xor(p, 2, 32);
            p += __shfl_xor(p, 4, 32);
            p += __shfl_xor(p, 8, 32);
            p += __shfl_xor(p, 16, 32);
            sc[m] = p;
        }
        float mx = sc[0];
#pragma unroll
        for (int m = 1; m <= 8; ++m) mx = fmaxf(mx, sc[m]);
        float e[9];
        float s = 0.f;
#pragma unroll
        for (int m = 0; m <= 8; ++m) { e[m] = __expf(sc[m] - mx); s += e[m]; }
        const float inv = 1.0f / s;
        if (lane == 0) {
#pragma unroll
            for (int m = 0; m <= 8; ++m) attnS[wave][r * 9 + m] = e[m] * inv;
        }
    }
    __syncthreads();

    // ---- Phase 5: adjusted = attn[:,0]*g_label + scatter(attn[:,0:K], labels) ----
    // Emitted straight from the WMMA register layout of accP (= g_label).
    // Faithful to the reference: scatter uses attention columns 0..K-1.
#pragma unroll
    for (int j = 0; j < 8; ++j) {
        const int M    = j + half * 8;
        const int rowG = rowBase + M;
        const int4* lb = reinterpret_cast<const int4*>(labels + (size_t)rowG * KRET);
        const int4 lab0 = lb[0];
        const int4 lab1 = lb[1];
        const int  labv[8] = {lab0.x, lab0.y, lab0.z, lab0.w,
                              lab1.x, lab1.y, lab1.z, lab1.w};
        float ak[8];
#pragma unroll
        for (int k = 0; k < 8; ++k) ak[k] = attnS[wave][M * 9 + k];
        const float a0 = ak[0];   // attention[:, :1]
#pragma unroll
        for (int n = 0; n < 8; ++n) {
            const int c = n * 16 + l15;
            float v = a0 * accP[n][j];
#pragma unroll
            for (int k = 0; k < 8; ++k) v += (labv[k] == c) ? ak[k] : 0.f;
            out[(size_t)rowG * CDIM + c] = v;
        }
    }
}

extern "C" void kernel_launch(void* const* d_in, const int* in_sizes, int n_in,
                              void* d_out, int out_size, void* d_ws, size_t ws_size,
                              hipStream_t stream) {
    const float* g      = (const float*)d_in[0];  // graph_embeddings  [B, D]
    const float* retr   = (const float*)d_in[1];  // retrieval_embeddings [B, K, D]
    const float* Wa     = (const float*)d_in[2];  // W_adapter [D, D]
    const float* Wp     = (const float*)d_in[3];  // W_pred    [D, C]
    const float* bp     = (const float*)d_in[4];  // b_pred    [C]
    const int*   labels = (const int*)d_in[5];    // labels    [B, K]
    float* out = (float*)d_out;                   // [B, C]

    dim3 grid(B_ROWS / ROWS_PER_BLOCK);
    dim3 block(WAVES * 32);
    graph_retrieval_kernel<<<grid, block, 0, stream>>>(g, retr, Wa, Wp, bp, labels, out);
}